// ClusteredAttention_71347996721627
// MI455X (gfx1250) — compile-verified
//
#include <hip/hip_runtime.h>
#include <hip/hip_bf16.h>

typedef float v2f __attribute__((ext_vector_type(2)));
typedef float v8f __attribute__((ext_vector_type(8)));
typedef int   v4i __attribute__((ext_vector_type(4)));

typedef __attribute__((address_space(1))) v4i gv4i;   // global b128 chunk
typedef __attribute__((address_space(3))) v4i lv4i;   // LDS b128 chunk

#define B_   2
#define L_   512
#define V_   64
#define D_   128
#define NJT  (L_ / 16)   // 32 j-tiles
#define NIT  (L_ / 16)   // 32 i-tiles
#define WPB  4           // waves per block (wave32)

#if defined(__gfx1250__) && __has_builtin(__builtin_amdgcn_global_load_async_to_lds_b128)
#define USE_ASYNC 1
#else
#define USE_ASYNC 0
#endif

// ---------------- Stage 1: sumT[b,j,d] = sum_v key[b,j,v,d] ----------------
__global__ __launch_bounds__(256) void ca_sumkey_kernel(
    const float* __restrict__ key, float* __restrict__ sumt)
{
    int idx = blockIdx.x * 256 + threadIdx.x;   // over B*L*D = 131072, exact
    int d   = idx & (D_ - 1);
    int bl  = idx >> 7;                          // b*L + j
    const float* kp = key + (size_t)bl * V_ * D_ + d;
    float s = 0.0f;
#pragma unroll 8
    for (int v = 0; v < V_; ++v) s += kp[v * D_];
    sumt[idx] = s;
}

// ---------------- Stage 2: fused clustered flash attention ----------------
// grid: B*V*NIT/WPB blocks of WPB*32 threads; wave w -> (b, k, i-tile)
__global__ __launch_bounds__(WPB * 32) void ca_attn_kernel(
    const float* __restrict__ q,
    const float* __restrict__ value,
    const int*   __restrict__ label,
    const float* __restrict__ sumt,
    float*       __restrict__ out)
{
#if USE_ASYNC
    __shared__ float stile[2][16 * D_];       // double-buffered sumT tile
    __shared__ float vtile[2][16 * D_];       // double-buffered value tile
#else
    __shared__ float stile[1][16 * D_];
    __shared__ float vtile[1][16 * D_];
#endif
    __shared__ float ptile[WPB][16 * 16];     // per-wave P scratch (m, n)

    const int tid  = threadIdx.x;
    const int widx = tid >> 5;
    const int lane = tid & 31;
    const int lo   = lane & 15;    // M (rows) for A, N (cols) for B/C/D
    const int hi   = lane >> 4;    // half-wave select

    const int w  = blockIdx.x * WPB + widx;
    const int it = w & (NIT - 1);
    const int k  = (w >> 5) & (V_ - 1);
    const int b  = w >> 11;
    const int i0 = it * 16;

    // ---- preload Q as WMMA-A fragments: A[M=lo][K=4c+2hi+r], 32 chunks ----
    v2f qa[32];
    {
        const float* qrow = q + (((size_t)(b * L_ + i0 + lo)) * V_ + k) * D_ + 2 * hi;
#pragma unroll
        for (int c = 0; c < 32; ++c)
            qa[c] = *(const v2f*)(qrow + 4 * c);
    }

    // labels of the 8 i-rows this half-wave owns (m = r + 8*hi)
    int labi[8];
#pragma unroll
    for (int r = 0; r < 8; ++r)
        labi[r] = label[b * L_ + i0 + r + 8 * hi];

    v8f acc[8] = {};           // O: 16 x 128 fp32 accumulator (8 d-tiles)
    float mrow[8], lrow[8];
#pragma unroll
    for (int r = 0; r < 8; ++r) { mrow[r] = -1e30f; lrow[r] = 0.0f; }

    const float scale = 0.08838834764831845f;   // 1/sqrt(128)

    // stage tile jt into LDS buffer bsel
    auto stage = [&](int jt, int bsel) {
        const float* sp = sumt  + ((size_t)b * L_ + jt * 16) * D_;
        const float* vp = value + (((size_t)(b * L_ + jt * 16)) * V_ + k) * D_;
#if USE_ASYNC
        // 512 x 16B chunks per 8KB tile; 4 per thread per tile -> 8 async ops/wave
#pragma unroll
        for (int ii = 0; ii < 4; ++ii) {
            int c4 = tid + 128 * ii;           // chunk id 0..511
            int jl = c4 >> 5;
            int dc = (c4 & 31) * 4;
            __builtin_amdgcn_global_load_async_to_lds_b128(
                (gv4i*)(sp + c4 * 4), (lv4i*)&stile[bsel][c4 * 4], 0, 0);
            __builtin_amdgcn_global_load_async_to_lds_b128(
                (gv4i*)(vp + (size_t)jl * V_ * D_ + dc),
                (lv4i*)&vtile[bsel][jl * 128 + dc], 0, 0);
        }
#else
#pragma unroll
        for (int ii = 0; ii < 16; ++ii) {
            int idx = tid + 128 * ii;          // 2048 floats, coalesced
            int jl  = idx >> 7;
            int d   = idx & 127;
            stile[0][idx] = sp[idx];           // sumT rows are contiguous
            vtile[0][idx] = vp[(size_t)jl * V_ * D_ + d];
        }
        if (jt + 1 < NJT) {                    // hint next tiles -> global_prefetch_b8
            int jl = tid & 15;
            __builtin_prefetch(sp + 16 * D_ + jl * D_, 0, 0);
            __builtin_prefetch(vp + (size_t)(16 + jl) * V_ * D_, 0, 0);
        }
#endif
    };

#if USE_ASYNC
    stage(0, 0);                               // prologue fill of buffer 0
#endif

    for (int jt = 0; jt < NJT; ++jt) {
#if USE_ASYNC
        const int cur = jt & 1;
        if (jt + 1 < NJT) {
            stage(jt + 1, cur ^ 1);            // overlap next tile's DMA
            // 8 new ops may remain in flight; in-order completion => tile cur landed
            asm volatile("s_wait_asynccnt 0x8" ::: "memory");
        } else {
            asm volatile("s_wait_asynccnt 0x0" ::: "memory");
        }
#else
        const int cur = 0;
        stage(jt, 0);
#endif
        __syncthreads();                       // all waves' staging complete

        // ---- scores S = Q (16xD) x sumT_tile^T (Dx16), K in chunks of 4 ----
        v8f s = {};
#pragma unroll
        for (int c = 0; c < 32; ++c) {
            v2f bt = *(const v2f*)&stile[cur][lo * D_ + 4 * c + 2 * hi]; // B[K=d][N=j]
            s = __builtin_amdgcn_wmma_f32_16x16x4_f32(
                    false, qa[c], false, bt, (short)0, s, false, false);
        }

        // ---- mask + online softmax (rows m = r + 8*hi, cols n = lo) ----
        const int labj = label[b * L_ + jt * 16 + lo];
#pragma unroll
        for (int r = 0; r < 8; ++r) {
            float logit = (labi[r] == labj) ? s[r] * scale : -1e30f;
            float rmax = logit;
            rmax = fmaxf(rmax, __shfl_xor(rmax, 1));
            rmax = fmaxf(rmax, __shfl_xor(rmax, 2));
            rmax = fmaxf(rmax, __shfl_xor(rmax, 4));
            rmax = fmaxf(rmax, __shfl_xor(rmax, 8));   // per-16-lane-half row max
            float mnew  = fmaxf(mrow[r], rmax);
            float alpha = __expf(mrow[r] - mnew);
            float pe    = __expf(logit - mnew);
            float rsum  = pe;
            rsum += __shfl_xor(rsum, 1);
            rsum += __shfl_xor(rsum, 2);
            rsum += __shfl_xor(rsum, 4);
            rsum += __shfl_xor(rsum, 8);
            lrow[r] = lrow[r] * alpha + rsum;
            mrow[r] = mnew;
#pragma unroll
            for (int nt = 0; nt < 8; ++nt) acc[nt][r] *= alpha;
            ptile[widx][(r + 8 * hi) * 16 + lo] = pe;  // P[m][n] -> LDS
        }
        asm volatile("s_wait_dscnt 0" ::: "memory");   // P stores visible for re-read

        // ---- O += P (16x16) x Vtile (16x128); K=j in 4 chunks of 4 ----
#pragma unroll
        for (int cp = 0; cp < 4; ++cp) {
            // A[M=lo][K=4cp+2hi+r] from P scratch
            v2f pa = *(const v2f*)&ptile[widx][lo * 16 + 4 * cp + 2 * hi];
#pragma unroll
            for (int nt = 0; nt < 8; ++nt) {
                v2f vb;   // B[K=j_local][N=d]
                vb.x = vtile[cur][(4 * cp + 2 * hi + 0) * D_ + nt * 16 + lo];
                vb.y = vtile[cur][(4 * cp + 2 * hi + 1) * D_ + nt * 16 + lo];
                acc[nt] = __builtin_amdgcn_wmma_f32_16x16x4_f32(
                              false, pa, false, vb, (short)0, acc[nt], false, false);
            }
        }
        __syncthreads();    // all waves done with buffer `cur` before it is refilled
    }

    // ---- finalize: out[b, i0+m, k, d] = O[m][d] / l[m] ----
#pragma unroll
    for (int r = 0; r < 8; ++r) {
        float inv = 1.0f / lrow[r];
        int   m   = r + 8 * hi;
        size_t base = (((size_t)(b * L_ + i0 + m)) * V_ + k) * D_;
#pragma unroll
        for (int nt = 0; nt < 8; ++nt)
            out[base + nt * 16 + lo] = acc[nt][r] * inv;
    }
}

extern "C" void kernel_launch(void* const* d_in, const int* in_sizes, int n_in,
                              void* d_out, int out_size, void* d_ws, size_t ws_size,
                              hipStream_t stream) {
    const float* q     = (const float*)d_in[0];
    const float* key   = (const float*)d_in[1];
    const float* value = (const float*)d_in[2];
    const int*   label = (const int*)d_in[3];
    float*       out   = (float*)d_out;
    float*       sumt  = (float*)d_ws;          // B*L*D floats = 512 KB

    // Stage 1: key reduction over v
    ca_sumkey_kernel<<<(B_ * L_ * D_) / 256, 256, 0, stream>>>(key, sumt);

    // Stage 2: fused masked-softmax attention, 1 wave per (b,k,i-tile)
    const int nwaves = B_ * V_ * NIT;           // 4096
    ca_attn_kernel<<<nwaves / WPB, WPB * 32, 0, stream>>>(q, value, label, sumt, out);
}